// AtomAttentionEncoder_36172214567336
// MI455X (gfx1250) — compile-verified
//
#include <hip/hip_runtime.h>

// ---------------- CDNA5 WMMA types ----------------
typedef __attribute__((ext_vector_type(2))) float v2f;
typedef __attribute__((ext_vector_type(8))) float v8f;

// ---------------------------------------------------------------------------
// Generic f32 WMMA GEMM: D = act(A[M,K] @ B[K,N] (+ Cin)), one wave per 16x16
// tile, K stepped by 4 with V_WMMA_F32_16X16X4_F32. All behavioral flags are
// template parameters so the K-loop is pure load+WMMA (no cndmask/branches).
//   ACT: 0 = none, 1 = relu, 2 = sigmoid
//   RELU_A: apply relu to A elements as loaded (for relu(c) @ W_cond)
// A/B/C/D layouts per CDNA5 ISA 7.12.2:
//   A 16x4 : lane&15 = row M, lane>=16 holds K+2/K+3 (2 VGPRs)
//   B 4x16 : lane&15 = col N, lane>=16 holds K+2/K+3
//   C/D    : vgpr r -> row r (lanes 0-15) / row r+8 (lanes 16-31), col = lane&15
// ---------------------------------------------------------------------------
template <int ACT, bool RELU_A, bool HAS_CIN>
__global__ __launch_bounds__(256) void gemm_wmma_f32(
    const float* __restrict__ A, int lda,
    const float* __restrict__ B, int ldb,
    const float* __restrict__ Cin,
    float* __restrict__ D,
    int M, int N, int K) {
  int lane = threadIdx.x & 31;
  int wid  = (blockIdx.x * blockDim.x + threadIdx.x) >> 5;
  int tilesN   = N >> 4;
  int numTiles = (M >> 4) * tilesN;
  if (wid >= numTiles) return;                 // uniform per wave
  int tm = (wid / tilesN) << 4;
  int tn = (wid % tilesN) << 4;
  int rsel = lane & 15;
  int koff = (lane >> 4) << 1;                 // 0 or 2
  const float* Arow = A + (size_t)(tm + rsel) * lda + koff;   // 8B aligned
  const float* Bcol = B + (size_t)koff * ldb + tn + rsel;
  // prefetch this lane's A row into the cache hierarchy (global_prefetch_b8)
  for (int pk = 0; pk < K; pk += 32) __builtin_prefetch(Arow + pk, 0, 1);
  v8f acc = {};
#pragma unroll 8
  for (int k = 0; k < K; k += 4) {
    v2f a = *(const v2f*)(Arow + k);           // single b64 load
    if (RELU_A) { a.x = fmaxf(a.x, 0.f); a.y = fmaxf(a.y, 0.f); }
    v2f b;
    b.x = Bcol[(size_t)k * ldb];
    b.y = Bcol[(size_t)k * ldb + ldb];
    acc = __builtin_amdgcn_wmma_f32_16x16x4_f32(false, a, false, b,
                                                (short)0, acc, false, false);
  }
  int rowbase = tm + ((lane >> 4) << 3);
#pragma unroll
  for (int rr = 0; rr < 8; ++rr) {
    float v = acc[rr];
    size_t idx = (size_t)(rowbase + rr) * N + tn + rsel;
    if constexpr (HAS_CIN) v += Cin[idx];
    if constexpr (ACT == 1) v = fmaxf(v, 0.f);
    if constexpr (ACT == 2) v = 1.f / (1.f + __expf(-v));
    D[idx] = v;
  }
}

// ---------------------------------------------------------------------------
// Wave-per-row LayerNorm: dst[i] = LN(src[i % src_mod]) * g + b, width mult of 32
// ---------------------------------------------------------------------------
__global__ void ln_rows_kernel(const float* __restrict__ src, float* __restrict__ dst,
                               const float* __restrict__ g, const float* __restrict__ b,
                               int rows, int width, int src_mod) {
  int wid  = (blockIdx.x * blockDim.x + threadIdx.x) >> 5;
  int lane = threadIdx.x & 31;
  if (wid >= rows) return;
  const float* row = src + (size_t)(wid % src_mod) * width;
  int per = width >> 5;
  float vals[12];
  float s = 0.f;
  for (int e = 0; e < per; ++e) { vals[e] = row[e * 32 + lane]; s += vals[e]; }
  for (int o = 16; o; o >>= 1) s += __shfl_xor(s, o, 32);
  float mean = s / (float)width;
  float var = 0.f;
  for (int e = 0; e < per; ++e) { float d = vals[e] - mean; var += d * d; }
  for (int o = 16; o; o >>= 1) var += __shfl_xor(var, o, 32);
  float rstd = rsqrtf(var / (float)width + 1e-5f);
  float* orow = dst + (size_t)wid * width;
  for (int e = 0; e < per; ++e) {
    int cidx = e * 32 + lane;
    orow[cidx] = (vals[e] - mean) * rstd * g[cidx] + b[cidx];
  }
}

// ---------------------------------------------------------------------------
// zp[ti,tj,:16] = LN(z[ti,tj,:128]) @ W_pair  (computed ONCE per token pair,
// reused for all 9 tiled (i,j) atom pairs). Wave per z row.
// ---------------------------------------------------------------------------
__global__ void zp_kernel(const float* __restrict__ z,
                          const float* __restrict__ g, const float* __restrict__ b,
                          const float* __restrict__ W_pair, float* __restrict__ zp) {
  int wid  = (blockIdx.x * blockDim.x + threadIdx.x) >> 5;
  int lane = threadIdx.x & 31;
  if (wid >= 256 * 256) return;
  const float* row = z + (size_t)wid * 128;
  float4 v = *(const float4*)(row + lane * 4);
  float s = v.x + v.y + v.z + v.w;
  for (int o = 16; o; o >>= 1) s += __shfl_xor(s, o, 32);
  float mean = s * (1.f / 128.f);
  float d0 = v.x - mean, d1 = v.y - mean, d2 = v.z - mean, d3 = v.w - mean;
  float vv = d0 * d0 + d1 * d1 + d2 * d2 + d3 * d3;
  for (int o = 16; o; o >>= 1) vv += __shfl_xor(vv, o, 32);
  float rstd = rsqrtf(vv * (1.f / 128.f) + 1e-5f);
  int k0 = lane * 4;
  float n0 = d0 * rstd * g[k0 + 0] + b[k0 + 0];
  float n1 = d1 * rstd * g[k0 + 1] + b[k0 + 1];
  float n2 = d2 * rstd * g[k0 + 2] + b[k0 + 2];
  float n3 = d3 * rstd * g[k0 + 3] + b[k0 + 3];
  float res = 0.f;
  for (int c = 0; c < 16; ++c) {
    float p = n0 * W_pair[(k0 + 0) * 16 + c] + n1 * W_pair[(k0 + 1) * 16 + c] +
              n2 * W_pair[(k0 + 2) * 16 + c] + n3 * W_pair[(k0 + 3) * 16 + c];
    for (int o = 16; o; o >>= 1) p += __shfl_xor(p, o, 32);
    if (lane == c) res = p;
  }
  if (lane < 16) zp[(size_t)wid * 16 + lane] = res;
}

// ---------------------------------------------------------------------------
// pos_c[i] = ref_pos[i] @ W_pos ; x0[i] = pos_c[i] + r[i] @ W_noisy
// ---------------------------------------------------------------------------
__global__ void pos_init_kernel(const float* __restrict__ ref_pos,
                                const float* __restrict__ r,
                                const float* __restrict__ W_pos,
                                const float* __restrict__ W_noisy,
                                float* __restrict__ pos_c, float* __restrict__ x) {
  int i = blockIdx.x, ch = threadIdx.x;
  float px = ref_pos[i * 3 + 0], py = ref_pos[i * 3 + 1], pz = ref_pos[i * 3 + 2];
  float pc = px * W_pos[ch] + py * W_pos[128 + ch] + pz * W_pos[256 + ch];
  pos_c[(size_t)i * 128 + ch] = pc;
  float rx = r[i * 3 + 0], ry = r[i * 3 + 1], rz = r[i * 3 + 2];
  x[(size_t)i * 128 + ch] = pc + rx * W_noisy[ch] + ry * W_noisy[128 + ch] + rz * W_noisy[256 + ch];
}

// ---------------------------------------------------------------------------
// pair[i,j,:] = mask*(dist@W_off + inv_sq*W_inv + W_mask) + zp[i%256,j%256]
//             + conda[i] + condb[j], then += 3-layer relu MLP(16x16) of itself.
// One thread per (i,j); the three 16x16 weights live in LDS. Single pass that
// writes the 37.7MB pair tensor exactly once.
// ---------------------------------------------------------------------------
__global__ void pair_build_kernel(const float* __restrict__ ref_pos, const int* __restrict__ uid,
                                  const float* __restrict__ W_off, const float* __restrict__ W_inv,
                                  const float* __restrict__ W_mask,
                                  const float* __restrict__ zp,
                                  const float* __restrict__ conda, const float* __restrict__ condb,
                                  const float* __restrict__ W1, const float* __restrict__ W2,
                                  const float* __restrict__ W3,
                                  float* __restrict__ pair) {
  __shared__ float sW1[256], sW2[256], sW3[256], sOff[48], sInv[16], sMask[16];
  int tid = threadIdx.x;
  if (tid < 256) { sW1[tid] = W1[tid]; sW2[tid] = W2[tid]; sW3[tid] = W3[tid]; }
  if (tid < 48) sOff[tid] = W_off[tid];
  if (tid < 16) { sInv[tid] = W_inv[tid]; sMask[tid] = W_mask[tid]; }
  __syncthreads();
  size_t t = (size_t)blockIdx.x * blockDim.x + tid;  // 768*768 threads
  int i = (int)(t / 768), j = (int)(t % 768);
  float dx = ref_pos[i * 3 + 0] - ref_pos[j * 3 + 0];
  float dy = ref_pos[i * 3 + 1] - ref_pos[j * 3 + 1];
  float dz = ref_pos[i * 3 + 2] - ref_pos[j * 3 + 2];
  float m  = (uid[i] == uid[j]) ? 1.f : 0.f;
  float inv = 1.f / (1.f + dx * dx + dy * dy + dz * dz);
  const float* zr = zp + ((size_t)(i % 256) * 256 + (j % 256)) * 16;
  const float* ca = conda + (size_t)i * 16;
  const float* cb = condb + (size_t)j * 16;
  float p[16];
#pragma unroll
  for (int c = 0; c < 16; ++c) {
    float geo = dx * sOff[c] + dy * sOff[16 + c] + dz * sOff[32 + c] + inv * sInv[c] + sMask[c];
    p[c] = m * geo + zr[c] + ca[c] + cb[c];
  }
  float h1[16], h2[16];
#pragma unroll
  for (int o = 0; o < 16; ++o) {
    float sm = 0.f;
    for (int c = 0; c < 16; ++c) sm += fmaxf(p[c], 0.f) * sW1[c * 16 + o];
    h1[o] = sm;
  }
#pragma unroll
  for (int o = 0; o < 16; ++o) {
    float sm = 0.f;
    for (int c = 0; c < 16; ++c) sm += fmaxf(h1[c], 0.f) * sW2[c * 16 + o];
    h2[o] = sm;
  }
  float* outp = pair + t * 16;
#pragma unroll
  for (int o = 0; o < 16; ++o) {
    float sm = 0.f;
    for (int c = 0; c < 16; ++c) sm += fmaxf(h2[c], 0.f) * sW3[c * 16 + o];
    outp[o] = p[o] + sm;
  }
}

// ---------------------------------------------------------------------------
// Local attention bias: only the 128-wide key window per query is needed.
// bias_loc[i, jj, h] = LN16(pair[i, j0(i)+jj]) @ Wb[:,h]  (0 for invalid j)
// ---------------------------------------------------------------------------
__global__ void bias_kernel(const float* __restrict__ pair,
                            const float* __restrict__ lnp_g, const float* __restrict__ lnp_b,
                            const float* __restrict__ Wb, float* __restrict__ bias_loc) {
  int t = blockIdx.x * blockDim.x + threadIdx.x;  // 768*128
  int i = t >> 7, jj = t & 127;
  int j = ((i >> 5) << 5) - 48 + jj;
  float4 out = {0.f, 0.f, 0.f, 0.f};
  if (j >= 0 && j < 768) {
    const float* p = pair + ((size_t)i * 768 + j) * 16;
    float v[16]; float s = 0.f;
#pragma unroll
    for (int c = 0; c < 16; ++c) { v[c] = p[c]; s += v[c]; }
    float mean = s * (1.f / 16.f);
    float var = 0.f;
#pragma unroll
    for (int c = 0; c < 16; ++c) { float d = v[c] - mean; var += d * d; }
    float rstd = rsqrtf(var * (1.f / 16.f) + 1e-5f);
#pragma unroll
    for (int c = 0; c < 16; ++c) {
      float nv = (v[c] - mean) * rstd * lnp_g[c] + lnp_b[c];
      out.x += nv * Wb[c * 4 + 0];
      out.y += nv * Wb[c * 4 + 1];
      out.z += nv * Wb[c * 4 + 2];
      out.w += nv * Wb[c * 4 + 3];
    }
  }
  *(float4*)(bias_loc + (size_t)t * 4) = out;
}

// ---------------------------------------------------------------------------
// Local attention, one wave per (query block of 32, head). All global/LDS
// loads are UNCONDITIONAL:
//  - K rows are address-clamped; garbage only lands in columns that get an
//    additive -1e30 mask before the LDS store (absorbed exactly in f32, and
//    exp underflows to +0 in softmax). The additive-mask form (instead of a
//    select) keeps LLVM from sinking the bias load under a divergent branch.
//  - Phase 2: softmax leaves P == +0 for invalid columns, so clamped V rows
//    contribute 0. Each (jk, jk+1) chunk is entirely valid or invalid.
// P padded to 130 floats/row: even stride -> aligned b64 LDS accesses,
// stride mod 64 == 2 -> conflict-free across 32 lanes.
// __launch_bounds__(32, 1): full VGPR budget, keeps Q fragments from spilling.
// ---------------------------------------------------------------------------
__global__ __launch_bounds__(32, 1) void attention_kernel(
    const float* __restrict__ qh, const float* __restrict__ kh,
    const float* __restrict__ vh, const float* __restrict__ gsig,
    const float* __restrict__ bias_loc, float* __restrict__ og) {
  __shared__ float P[32][130];
  int lane = threadIdx.x;            // blockDim == 32 (one wave)
  int head = blockIdx.x & 3;
  int qb   = blockIdx.x >> 2;
  int j0   = qb * 32 - 48;           // window start (AF3 local mask)
  int rsel = lane & 15;
  int hi   = lane >> 4;
  int koff = hi << 1;
  const float scale = 0.17677669529663687f;   // 1/sqrt(32)

  // -------- Phase 1: logits = Q K^T * scale + bias, masked, into LDS --------
  for (int mt = 0; mt < 2; ++mt) {
    int qrow = qb * 32 + mt * 16 + rsel;
    const float* Arow = qh + (size_t)qrow * 128 + head * 32 + koff;
    v2f afrag[8];
#pragma unroll
    for (int kk = 0; kk < 8; ++kk) afrag[kk] = *(const v2f*)(Arow + kk * 4);
    for (int nt = 0; nt < 8; ++nt) {
      int jc = j0 + nt * 16 + rsel;             // key index for this B column
      float maskadd = (jc >= 0 && jc < 768) ? 0.f : -1e30f;   // one cndmask
      int jcl = min(max(jc, 0), 767);           // clamped row, always loadable
      const float* Bbase = kh + (size_t)jcl * 128 + head * 32 + koff;
      v8f acc = {};
#pragma unroll
      for (int kk = 0; kk < 8; ++kk) {
        v2f b = *(const v2f*)(Bbase + kk * 4);  // single b64, unconditional
        acc = __builtin_amdgcn_wmma_f32_16x16x4_f32(false, afrag[kk], false, b,
                                                    (short)0, acc, false, false);
      }
      int jj = nt * 16 + rsel;                  // 0..127 window offset
      const float* brow = bias_loc + ((size_t)(qb * 32 + mt * 16 + hi * 8) * 128 + jj) * 4 + head;
#pragma unroll
      for (int rr = 0; rr < 8; ++rr) {
        int qr = mt * 16 + hi * 8 + rr;         // row within the 32-query block
        float bv = brow[(size_t)rr * 512];      // unconditional, in-bounds
        P[qr][jj] = fmaf(acc[rr], scale, bv) + maskadd;
      }
    }
  }
  __syncthreads();
  // -------- softmax: lane owns one query row (b64 LDS, conflict-free) ------
  {
    float* prow = &P[lane][0];
    float mx = -3e38f;
#pragma unroll 8
    for (int k = 0; k < 128; k += 2) {
      v2f t = *(const v2f*)(prow + k);
      mx = fmaxf(mx, fmaxf(t.x, t.y));
    }
    float sum = 0.f;
#pragma unroll 8
    for (int k = 0; k < 128; k += 2) {
      v2f t = *(v2f*)(prow + k);
      t.x = __expf(t.x - mx); t.y = __expf(t.y - mx);
      sum += t.x + t.y;
      *(v2f*)(prow + k) = t;
    }
    float invs = 1.f / sum;
#pragma unroll 8
    for (int k = 0; k < 128; k += 2) {
      v2f t = *(v2f*)(prow + k);
      t.x *= invs; t.y *= invs;
      *(v2f*)(prow + k) = t;
    }
  }
  __syncthreads();
  // -------- Phase 2: O = P @ V, gated ---------------------------------------
  for (int mt = 0; mt < 2; ++mt) {
    const float* Pr = &P[mt * 16 + rsel][koff];
    for (int dt = 0; dt < 2; ++dt) {
      const float* Vcol = vh + head * 32 + dt * 16 + rsel;
      v8f acc = {};
#pragma unroll 8
      for (int kk = 0; kk < 32; ++kk) {
        v2f a = *(const v2f*)(Pr + kk * 4);     // b64 from LDS
        int jk = j0 + kk * 4 + koff;            // even; chunk fully (in)valid
        int jkc = ((unsigned)jk > 766u) ? 0 : jk;  // clamp, P==0 kills garbage
        const float* vb = Vcol + (size_t)jkc * 128;
        v2f b; b.x = vb[0]; b.y = vb[128];
        acc = __builtin_amdgcn_wmma_f32_16x16x4_f32(false, a, false, b,
                                                    (short)0, acc, false, false);
      }
#pragma unroll
      for (int rr = 0; rr < 8; ++rr) {
        int qabs = qb * 32 + mt * 16 + hi * 8 + rr;
        int d = head * 32 + dt * 16 + rsel;
        size_t idx = (size_t)qabs * 128 + d;
        og[idx] = acc[rr] * gsig[idx];
      }
    }
  }
}

// a_out[t] = mean over the 3 atoms of token t of qout
__global__ void mean_kernel(const float* __restrict__ qout, float* __restrict__ a_out) {
  int t = blockIdx.x * blockDim.x + threadIdx.x;   // 256*384
  int tok = t / 384, ct = t % 384;
  float v = qout[(size_t)(tok * 3 + 0) * 384 + ct] +
            qout[(size_t)(tok * 3 + 1) * 384 + ct] +
            qout[(size_t)(tok * 3 + 2) * 384 + ct];
  a_out[t] = v * (1.f / 3.f);
}

// ---------------------------------------------------------------------------
static inline void launch_gemm(const float* A, int lda, const float* B, int ldb,
                               const float* Cin, float* D, int M, int N, int K,
                               int act, int relu_a, hipStream_t s) {
  int tiles = (M / 16) * (N / 16);   // all call sites: tiles % 8 == 0
  dim3 g(tiles / 8), blk(256);
  if (relu_a)
    gemm_wmma_f32<0, true, false><<<g, blk, 0, s>>>(A, lda, B, ldb, nullptr, D, M, N, K);
  else if (Cin)
    gemm_wmma_f32<0, false, true><<<g, blk, 0, s>>>(A, lda, B, ldb, Cin, D, M, N, K);
  else if (act == 1)
    gemm_wmma_f32<1, false, false><<<g, blk, 0, s>>>(A, lda, B, ldb, nullptr, D, M, N, K);
  else if (act == 2)
    gemm_wmma_f32<2, false, false><<<g, blk, 0, s>>>(A, lda, B, ldb, nullptr, D, M, N, K);
  else
    gemm_wmma_f32<0, false, false><<<g, blk, 0, s>>>(A, lda, B, ldb, nullptr, D, M, N, K);
}

extern "C" void kernel_launch(void* const* d_in, const int* in_sizes, int n_in,
                              void* d_out, int out_size, void* d_ws, size_t ws_size,
                              hipStream_t stream) {
  (void)in_sizes; (void)n_in; (void)out_size; (void)ws_size;
  const float* ref_pos  = (const float*)d_in[0];
  const int*   uid      = (const int*)d_in[1];
  const float* r        = (const float*)d_in[2];
  const float* s        = (const float*)d_in[3];
  const float* z        = (const float*)d_in[4];
  const float* W_pos    = (const float*)d_in[5];
  const float* W_off    = (const float*)d_in[6];
  const float* W_inv    = (const float*)d_in[7];
  const float* W_mask   = (const float*)d_in[8];
  const float* W_ca     = (const float*)d_in[9];
  const float* W_cb     = (const float*)d_in[10];
  const float* ln_s_g   = (const float*)d_in[11];
  const float* ln_s_b   = (const float*)d_in[12];
  const float* W_single = (const float*)d_in[13];
  const float* ln_z_g   = (const float*)d_in[14];
  const float* ln_z_b   = (const float*)d_in[15];
  const float* W_pair   = (const float*)d_in[16];
  const float* W_noisy  = (const float*)d_in[17];
  const float* W_mlp1   = (const float*)d_in[18];
  const float* W_mlp2   = (const float*)d_in[19];
  const float* W_mlp3   = (const float*)d_in[20];
  const float* at_ln_g  = (const float*)d_in[21];
  const float* at_ln_b  = (const float*)d_in[22];
  const float* at_Wq    = (const float*)d_in[23];
  const float* at_Wk    = (const float*)d_in[24];
  const float* at_Wv    = (const float*)d_in[25];
  const float* at_Wg    = (const float*)d_in[26];
  const float* at_lnp_g = (const float*)d_in[27];
  const float* at_lnp_b = (const float*)d_in[28];
  const float* at_Wb    = (const float*)d_in[29];
  const float* at_Wo    = (const float*)d_in[30];
  const float* at_ln2_g = (const float*)d_in[31];
  const float* at_ln2_b = (const float*)d_in[32];
  const float* at_Wt1   = (const float*)d_in[33];
  const float* at_Wt2   = (const float*)d_in[34];
  const float* W_proj   = (const float*)d_in[35];

  float* out   = (float*)d_out;          // [a_out | qout | c | pair]
  float* a_out = out;
  float* qout  = out + 256 * 384;
  float* c_out = qout + 768 * 384;
  float* pair  = c_out + 768 * 128;

  float* ws = (float*)d_ws;
  size_t off = 0;
  auto alloc = [&](size_t n) { float* p = ws + off; off += n; return p; };
  float* pos_c = alloc(768 * 128);
  float* x     = alloc(768 * 128);
  float* snorm = alloc(768 * 384);
  float* zp    = alloc(65536 * 16);
  float* conda = alloc(768 * 16);
  float* condb = alloc(768 * 16);
  float* abuf  = alloc(768 * 128);
  float* qbuf  = alloc(768 * 128);
  float* kbuf  = alloc(768 * 128);
  float* vbuf  = alloc(768 * 128);
  float* gbuf  = alloc(768 * 128);
  float* ogbuf = alloc(768 * 128);
  float* tbuf  = alloc(768 * 128);
  float* hmid  = alloc(768 * 512);
  float* biasb = alloc(768 * 128 * 4);   // ~12.2 MB total workspace

  pos_init_kernel<<<768, 128, 0, stream>>>(ref_pos, r, W_pos, W_noisy, pos_c, x);
  ln_rows_kernel<<<96, 256, 0, stream>>>(s, snorm, ln_s_g, ln_s_b, 768, 384, 256);
  launch_gemm(snorm, 384, W_single, 128, pos_c, c_out, 768, 128, 384, 0, 0, stream);
  launch_gemm(c_out, 128, W_ca, 16, nullptr, conda, 768, 16, 128, 0, 1, stream);
  launch_gemm(c_out, 128, W_cb, 16, nullptr, condb, 768, 16, 128, 0, 1, stream);
  zp_kernel<<<8192, 256, 0, stream>>>(z, ln_z_g, ln_z_b, W_pair, zp);
  pair_build_kernel<<<2304, 256, 0, stream>>>(ref_pos, uid, W_off, W_inv, W_mask,
                                              zp, conda, condb, W_mlp1, W_mlp2, W_mlp3, pair);

  for (int b = 0; b < 3; ++b) {
    ln_rows_kernel<<<96, 256, 0, stream>>>(x, abuf, at_ln_g + b * 128, at_ln_b + b * 128, 768, 128, 768);
    launch_gemm(abuf, 128, at_Wq + b * 16384, 128, nullptr, qbuf, 768, 128, 128, 0, 0, stream);
    launch_gemm(abuf, 128, at_Wk + b * 16384, 128, nullptr, kbuf, 768, 128, 128, 0, 0, stream);
    launch_gemm(abuf, 128, at_Wv + b * 16384, 128, nullptr, vbuf, 768, 128, 128, 0, 0, stream);
    launch_gemm(abuf, 128, at_Wg + b * 16384, 128, nullptr, gbuf, 768, 128, 128, 2, 0, stream);
    bias_kernel<<<384, 256, 0, stream>>>(pair, at_lnp_g + b * 16, at_lnp_b + b * 16,
                                         at_Wb + b * 64, biasb);
    attention_kernel<<<96, 32, 0, stream>>>(qbuf, kbuf, vbuf, gbuf, biasb, ogbuf);
    launch_gemm(ogbuf, 128, at_Wo + b * 16384, 128, x, x, 768, 128, 128, 0, 0, stream);
    ln_rows_kernel<<<96, 256, 0, stream>>>(x, tbuf, at_ln2_g + b * 128, at_ln2_b + b * 128, 768, 128, 768);
    launch_gemm(tbuf, 128, at_Wt1 + b * 65536, 512, nullptr, hmid, 768, 512, 128, 1, 0, stream);
    launch_gemm(hmid, 512, at_Wt2 + b * 65536, 128, x, x, 768, 128, 512, 0, 0, stream);
  }

  launch_gemm(x, 128, W_proj, 384, nullptr, qout, 768, 384, 128, 1, 0, stream);
  mean_kernel<<<384, 256, 0, stream>>>(qout, a_out);
}